// GATLayer_56848187130165
// MI455X (gfx1250) — compile-verified
//
#include <hip/hip_runtime.h>

typedef __attribute__((ext_vector_type(2))) float v2f;
typedef __attribute__((ext_vector_type(8))) float v8f;

#define IN_CH   128
#define OCH     128      // HEADS * OUT_CH
#define HEADS   2
#define NEGSLP  0.2f

// ---------------- init ----------------
__global__ void gat_init_out(float* __restrict__ out, const float* __restrict__ bias, int total) {
  int t = blockIdx.x * blockDim.x + threadIdx.x;
  if (t < total) out[t] = bias[t & (OCH - 1)];
}

__global__ void gat_init_ms(unsigned* __restrict__ m_enc, float* __restrict__ ssum, int total) {
  int t = blockIdx.x * blockDim.x + threadIdx.x;
  if (t < total) { m_enc[t] = 0u; ssum[t] = 0.0f; }   // enc 0 < every finite float's encoding
}

// ---------------- GEMM: xp = x @ W via V_WMMA_F32_16X16X4_F32 ----------------
// One wave per 16x16 output tile; 8 waves (256 thr) per block cover all 8 column tiles
// of one 16-row strip. K loop: 128/4 = 32 WMMAs per tile.
__global__ void gat_gemm_wmma(const float* __restrict__ X, const float* __restrict__ W,
                              float* __restrict__ XP, int N) {
  const int lane = threadIdx.x & 31;
  const int wv   = threadIdx.x >> 5;      // 0..7 -> column tile
  const int half = lane >> 4;             // 0: lanes 0-15, 1: lanes 16-31
  const int l16  = lane & 15;
  const int m0   = blockIdx.x * 16;
  const int n0   = wv * 16;

  int arow = m0 + l16; if (arow >= N) arow = N - 1;     // clamp (N%16==0 normally)
  const float* __restrict__ xr = X + (long)arow * IN_CH;

  v8f c = {};
  for (int k0 = 0; k0 < IN_CH; k0 += 4) {
    // A 16x4: v0 = K=(0|2), v1 = K=(1|3) per half-wave; row = l16
    v2f a, b;
    a[0] = xr[k0 + 2 * half + 0];
    a[1] = xr[k0 + 2 * half + 1];
    // B 4x16: v0 = row K=(0|2), v1 = row K=(1|3); col = l16
    b[0] = W[(k0 + 2 * half + 0) * OCH + n0 + l16];
    b[1] = W[(k0 + 2 * half + 1) * OCH + n0 + l16];
    c = __builtin_amdgcn_wmma_f32_16x16x4_f32(false, a, false, b, (short)0, c, false, false);
  }
  // D 16x16 f32: VGPR r -> row m0 + r + 8*half, col n0 + l16
#pragma unroll
  for (int r = 0; r < 8; ++r) {
    int row = m0 + r + 8 * half;
    if (row < N) XP[(long)row * OCH + n0 + l16] = c[r];
  }
}

// ---------------- per-(node,head) attention coefficients ----------------
__global__ void gat_attn_coeff(const float* __restrict__ XP,
                               const float* __restrict__ att_src,
                               const float* __restrict__ att_dst,
                               float* __restrict__ a_src, float* __restrict__ a_dst, int N) {
  int t = blockIdx.x * blockDim.x + threadIdx.x;     // node*2 + head
  if (t >= N * HEADS) return;
  int h = t & 1, node = t >> 1;
  const float* __restrict__ p  = XP + (long)node * OCH + h * 64;
  const float* __restrict__ as = att_src + h * 64;
  const float* __restrict__ ad = att_dst + h * 64;
  float su = 0.f, du = 0.f;
#pragma unroll 4
  for (int cidx = 0; cidx < 64; cidx += 4) {
    float4 v = *(const float4*)(p + cidx);
    float4 s = *(const float4*)(as + cidx);
    float4 d = *(const float4*)(ad + cidx);
    su += v.x * s.x + v.y * s.y + v.z * s.z + v.w * s.w;
    du += v.x * d.x + v.y * d.y + v.z * d.z + v.w * d.w;
  }
  a_src[t] = su;
  a_dst[t] = du;
}

// monotonic f32 <-> u32 encoding for atomicMax-based float max
__device__ __forceinline__ unsigned f32_enc(float f) {
  unsigned b = __float_as_uint(f);
  return (b & 0x80000000u) ? ~b : (b | 0x80000000u);
}
__device__ __forceinline__ float f32_dec(unsigned e) {
  unsigned b = (e & 0x80000000u) ? (e & 0x7FFFFFFFu) : ~e;
  return __uint_as_float(b);
}

// ---------------- pass 1: alpha = leakyrelu(a_src[s]+a_dst[d]); segment max ----------------
__global__ void gat_edge_max(const int* __restrict__ ei,
                             const float* __restrict__ a_src, const float* __restrict__ a_dst,
                             float* __restrict__ alphaBuf, unsigned* __restrict__ m_enc,
                             int E, int Etot) {
  int t = blockIdx.x * blockDim.x + threadIdx.x;     // edge*2 + head
  if (t >= Etot * HEADS) return;
  int h = t & 1, e = t >> 1;
  int s, d;
  if (e < E) { s = ei[e]; d = ei[E + e]; } else { s = e - E; d = s; }
  float al = a_src[s * HEADS + h] + a_dst[d * HEADS + h];
  al = (al > 0.f) ? al : NEGSLP * al;                // leaky_relu
  alphaBuf[t] = al;
  atomicMax(&m_enc[d * HEADS + h], f32_enc(al));
}

// ---------------- pass 2: e = exp(alpha - m); segment sum ----------------
__global__ void gat_edge_expsum(const int* __restrict__ ei,
                                float* __restrict__ alphaBuf,
                                const unsigned* __restrict__ m_enc, float* __restrict__ ssum,
                                int E, int Etot) {
  int t = blockIdx.x * blockDim.x + threadIdx.x;
  if (t >= Etot * HEADS) return;
  int h = t & 1, e = t >> 1;
  int d = (e < E) ? ei[E + e] : (e - E);
  float m  = f32_dec(m_enc[d * HEADS + h]);
  float ex = expf(alphaBuf[t] - m);
  alphaBuf[t] = ex;                                  // reuse buffer: now holds exp term
  atomicAdd(&ssum[d * HEADS + h], ex);
}

// ---------------- pass 3: out[d] += (e/s) * xp[s]  (wave32 per edge) ----------------
// lane 0-15 -> head 0 channels [0,64), lane 16-31 -> head 1 channels [64,128);
// each lane handles 4 channels with a B128 load + 4 f32 global atomics (L2-resident out).
__global__ void gat_edge_scatter(const int* __restrict__ ei,
                                 const float* __restrict__ XP,
                                 const float* __restrict__ alphaBuf,
                                 const float* __restrict__ ssum,
                                 float* __restrict__ out, int E, int Etot) {
  int wave = blockIdx.x * (blockDim.x >> 5) + (threadIdx.x >> 5);
  if (wave >= Etot) return;                          // uniform per wave
  int lane = threadIdx.x & 31;
  int s, d;
  if (wave < E) { s = ei[wave]; d = ei[E + wave]; } else { s = wave - E; d = s; }
  int h  = lane >> 4;
  float w = alphaBuf[wave * HEADS + h] / (ssum[d * HEADS + h] + 1e-16f);
  int c0 = h * 64 + (lane & 15) * 4;
  float4 v = *(const float4*)(XP + (long)s * OCH + c0);
  float* o = out + (long)d * OCH + c0;
  atomicAdd(o + 0, w * v.x);
  atomicAdd(o + 1, w * v.y);
  atomicAdd(o + 2, w * v.z);
  atomicAdd(o + 3, w * v.w);
}

// ---------------- host ----------------
extern "C" void kernel_launch(void* const* d_in, const int* in_sizes, int n_in,
                              void* d_out, int out_size, void* d_ws, size_t ws_size,
                              hipStream_t stream) {
  const float* x        = (const float*)d_in[0];
  const int*   ei       = (const int*)d_in[1];     // [2, E] int32
  // d_in[2] = edge_weight: unused by reference attention math (edge_dim=None)
  const float* W        = (const float*)d_in[3];
  const float* att_src  = (const float*)d_in[4];
  const float* att_dst  = (const float*)d_in[5];
  const float* bias     = (const float*)d_in[6];
  float* out = (float*)d_out;

  const int N    = in_sizes[0] / IN_CH;
  const int E    = in_sizes[1] / 2;
  const int Etot = E + N;                           // with self loops

  // workspace layout (256B aligned slices)
  char* w = (char*)d_ws;
  auto align256 = [](size_t v) { return (v + 255) & ~(size_t)255; };
  size_t off = 0;
  float*    XP      = (float*)(w + off); off = align256(off + (size_t)N * OCH * 4);
  float*    a_src   = (float*)(w + off); off = align256(off + (size_t)N * HEADS * 4);
  float*    a_dst   = (float*)(w + off); off = align256(off + (size_t)N * HEADS * 4);
  unsigned* m_enc   = (unsigned*)(w + off); off = align256(off + (size_t)N * HEADS * 4);
  float*    ssum    = (float*)(w + off); off = align256(off + (size_t)N * HEADS * 4);
  float*    alphaB  = (float*)(w + off); off = align256(off + (size_t)Etot * HEADS * 4);
  (void)ws_size; (void)n_in; (void)out_size;

  const int TB = 256;

  // init: out = bias broadcast; m = -inf enc; s = 0
  gat_init_out<<<(N * OCH + TB - 1) / TB, TB, 0, stream>>>(out, bias, N * OCH);
  gat_init_ms <<<(N * HEADS + TB - 1) / TB, TB, 0, stream>>>(m_enc, ssum, N * HEADS);

  // dense projection via fp32 WMMA
  gat_gemm_wmma<<<(N + 15) / 16, TB, 0, stream>>>(x, W, XP, N);

  // per-node attention coefficients
  gat_attn_coeff<<<(N * HEADS + TB - 1) / TB, TB, 0, stream>>>(XP, att_src, att_dst, a_src, a_dst, N);

  // segment softmax over incoming edges (+ self loops)
  int eThreads = Etot * HEADS;
  gat_edge_max   <<<(eThreads + TB - 1) / TB, TB, 0, stream>>>(ei, a_src, a_dst, alphaB, m_enc, E, Etot);
  gat_edge_expsum<<<(eThreads + TB - 1) / TB, TB, 0, stream>>>(ei, alphaB, m_enc, ssum, E, Etot);

  // weighted message scatter: one wave per edge
  int wavesPerBlk = TB / 32;
  gat_edge_scatter<<<(Etot + wavesPerBlk - 1) / wavesPerBlk, TB, 0, stream>>>(ei, XP, alphaB, ssum, out, E, Etot);
}